// PaiConv_22273700397297
// MI455X (gfx1250) — compile-verified
//
#include <hip/hip_runtime.h>
#include <math.h>

#define BATCH 8
#define NPTS  2048
#define MTOT  (BATCH*NPTS)      // 16384
#define KNN   32
#define NNK   32
#define MAPS  32
#define INC   64
#define INCX  32
#define FCH   96
#define OUTC  64
#define FNN   (FCH*NNK)         // 3072

typedef float v2f __attribute__((ext_vector_type(2)));
typedef float v8f __attribute__((ext_vector_type(8)));

// ---------------------------------------------------------------------------
// Kernel 1: exact KNN (top-32 of the similarity row), one 256-thr block/query.
// pair(i,j) = -|xi|^2 + 2 xi.xj - |xj|^2 ; argmax, tie -> smaller index
// (matches jax.lax.top_k; self is always first with value 0).
// Wave-level shfl_xor argmax, then one 8-entry cross-wave LDS reduce.
// ---------------------------------------------------------------------------
__global__ __launch_bounds__(256) void knn_kernel(const float* __restrict__ x,
                                                  int* __restrict__ idx_out) {
    __shared__ float s_x[3][NPTS];   // 24 KB: whole batch's coords
    __shared__ float s_val[8];
    __shared__ int   s_ind[8];
    __shared__ int   s_win;

    const int m    = blockIdx.x;
    const int b    = m / NPTS;
    const int i    = m % NPTS;
    const int tid  = threadIdx.x;
    const int lane = tid & 31;
    const int wv   = tid >> 5;
    const float* xb = x + (size_t)b * 3 * NPTS;

    for (int t = tid; t < 3 * NPTS; t += 256)
        s_x[t / NPTS][t % NPTS] = xb[t];
    __syncthreads();

    const float xi0 = s_x[0][i], xi1 = s_x[1][i], xi2 = s_x[2][i];
    const float sqi = -(xi0 * xi0 + xi1 * xi1 + xi2 * xi2);
    const float NEG = -__builtin_inff();

    float v[8];
#pragma unroll
    for (int c = 0; c < 8; ++c) {
        const int j = tid + 256 * c;
        const float a0 = s_x[0][j], a1 = s_x[1][j], a2 = s_x[2][j];
        v[c] = sqi + 2.f * (xi0 * a0 + xi1 * a1 + xi2 * a2)
                   - (a0 * a0 + a1 * a1 + a2 * a2);
    }

    for (int r = 0; r < KNN; ++r) {
        float bv = NEG; int bj = 0x7fffffff;
#pragma unroll
        for (int c = 0; c < 8; ++c) {
            const int j = tid + 256 * c;
            if (v[c] > bv || (v[c] == bv && j < bj)) { bv = v[c]; bj = j; }
        }
        // wave32 argmax (no barriers)
#pragma unroll
        for (int off = 16; off > 0; off >>= 1) {
            const float ov = __shfl_xor(bv, off, 32);
            const int   oj = __shfl_xor(bj, off, 32);
            if (ov > bv || (ov == bv && oj < bj)) { bv = ov; bj = oj; }
        }
        if (lane == 0) { s_val[wv] = bv; s_ind[wv] = bj; }
        __syncthreads();
        if (tid == 0) {
            float fb = s_val[0]; int fj = s_ind[0];
#pragma unroll
            for (int w = 1; w < 8; ++w) {
                const float ov = s_val[w]; const int oj = s_ind[w];
                if (ov > fb || (ov == fb && oj < fj)) { fb = ov; fj = oj; }
            }
            s_win = fj;
            idx_out[(size_t)m * KNN + r] = fj;
        }
        __syncthreads();
        const int w = s_win;
        if ((w & 255) == tid) v[w >> 8] = NEG;   // knock out winner
        // next write to s_val/s_ind happens after every thread read s_win
    }
}

// ---------------------------------------------------------------------------
// Kernel 2: fused per-point pipeline, one 128-thread block (4 waves) / point.
//   geometry -> Fourier features -> MLP via WMMA -> perm (top-3 softmax)
//   -> shuffled feature gather -> WMMA: g = feats(96x32)@perm(32x32)
//   STORE_PHI=1: write g (phi row, 3072 floats) to workspace (for conv GEMM).
//   STORE_PHI=0: fused per-point GEMV against conv_w (fallback, small ws).
// ---------------------------------------------------------------------------
template <bool STORE_PHI>
__global__ __launch_bounds__(128) void point_kernel(
    const float* __restrict__ x, const float* __restrict__ feature,
    const float* __restrict__ B_mat, const float* __restrict__ kernels,
    const float* __restrict__ mlp_w, const float* __restrict__ mlp_b,
    const float* __restrict__ conv_w, const float* __restrict__ conv_b,
    const int* __restrict__ idx_in, float* __restrict__ phi,
    float* __restrict__ out) {

    __shared__ int   s_idx[KNN];
    __shared__ float s_xn[KNN][3];
    __shared__ float s_rel[KNN][3];
    __shared__ float s_dis[KNN];
    __shared__ float s_ff[KNN][2 * MAPS];        // 8 KB  (B of MLP WMMA)
    __shared__ float s_xf[INCX][KNN];            // 4 KB
    __shared__ float s_p[KNN][NNK];              // 4 KB
    __shared__ float s_perm[KNN][NNK + 1];       // padded stride 33
    __shared__ float s_feats[FCH][KNN + 1];      // padded stride 33
    __shared__ float s_g[FCH][KNN + 1];          // padded stride 33
    __shared__ float s_red[128];

    const int m    = blockIdx.x;
    const int b    = m / NPTS;
    const int i    = m % NPTS;
    const int tid  = threadIdx.x;
    const int lane = tid & 31;
    const int wave = tid >> 5;
    const int lm   = lane & 15;      // row/col within 16-wide tile
    const int kh   = lane >> 4;      // K-pair half

    if (tid < KNN) s_idx[tid] = idx_in[(size_t)m * KNN + tid];
    __syncthreads();

    if (tid < KNN) {
        const int j = s_idx[tid];
        const float* xb = x + (size_t)b * 3 * NPTS;
        s_xn[tid][0] = xb[0 * NPTS + j];
        s_xn[tid][1] = xb[1 * NPTS + j];
        s_xn[tid][2] = xb[2 * NPTS + j];
    }
    __syncthreads();
    if (tid < KNN) {
        const float r0 = s_xn[tid][0] - s_xn[0][0];
        const float r1 = s_xn[tid][1] - s_xn[0][1];
        const float r2 = s_xn[tid][2] - s_xn[0][2];
        s_rel[tid][0] = r0; s_rel[tid][1] = r1; s_rel[tid][2] = r2;
        s_dis[tid] = sqrtf(r0 * r0 + r1 * r1 + r2 * r2);
    }
    __syncthreads();

    // Fourier features: ff = [sin|cos](2*pi * [rep, rel, dis] @ B_mat)
#pragma unroll
    for (int r = 0; r < 8; ++r) {
        const int pid = tid + 128 * r;
        const int k = pid >> 5, j = pid & 31;
        float ph = s_xn[0][0]  * B_mat[0 * 32 + j]
                 + s_xn[0][1]  * B_mat[1 * 32 + j]
                 + s_xn[0][2]  * B_mat[2 * 32 + j]
                 + s_rel[k][0] * B_mat[3 * 32 + j]
                 + s_rel[k][1] * B_mat[4 * 32 + j]
                 + s_rel[k][2] * B_mat[5 * 32 + j]
                 + s_dis[k]    * B_mat[6 * 32 + j];
        ph *= 6.283185307179586f;
        s_ff[k][j]      = sinf(ph);
        s_ff[k][j + 32] = cosf(ph);
    }
    // p[k][n] = rel . kernels[:,n] (+ one_pad at [0][0])
#pragma unroll
    for (int r = 0; r < 8; ++r) {
        const int pid = tid + 128 * r;
        const int k = pid >> 5, n = pid & 31;
        float pv = s_rel[k][0] * kernels[0 * 32 + n]
                 + s_rel[k][1] * kernels[1 * 32 + n]
                 + s_rel[k][2] * kernels[2 * 32 + n];
        if (k == 0 && n == 0) pv += 1.0f;
        s_p[k][n] = pv;
    }
    __syncthreads();

    // ---- MLP via WMMA f32 16x16x4: x_feats[32,32] = mlp_w[32,64] @ ff^T ----
    {
        const int mt = wave >> 1, nt = wave & 1;   // 2x2 tiles over 4 waves
        const float* arow = mlp_w + (size_t)(mt * 16 + lm) * 64;
        v8f acc = {0.f, 0.f, 0.f, 0.f, 0.f, 0.f, 0.f, 0.f};
#pragma unroll
        for (int kk = 0; kk < 16; ++kk) {
            const int k0 = kk * 4 + kh * 2;
            v2f a, bb;
            a.x  = arow[k0];
            a.y  = arow[k0 + 1];
            bb.x = s_ff[nt * 16 + lm][k0];       // B[j][kpt] = ff[kpt][j]
            bb.y = s_ff[nt * 16 + lm][k0 + 1];
            acc = __builtin_amdgcn_wmma_f32_16x16x4_f32(
                      false, a, false, bb, (short)0, acc, false, false);
        }
#pragma unroll
        for (int r = 0; r < 8; ++r) {
            const int o = mt * 16 + 8 * kh + r;
            s_xf[o][nt * 16 + lm] = acc[r] + mlp_b[o];
        }
    }
    // topkmax: per column n, keep top-3 over k, softmax over k
    if (tid < NNK) {
        const int n = tid;
        float m1 = -3.4e38f, m2 = -3.4e38f, m3 = -3.4e38f;
        for (int k = 0; k < KNN; ++k) {
            const float pv = s_p[k][n];
            if (pv > m1)      { m3 = m2; m2 = m1; m1 = pv; }
            else if (pv > m2) { m3 = m2; m2 = pv; }
            else if (pv > m3) { m3 = pv; }
        }
        float sum = 0.f;
        for (int k = 0; k < KNN; ++k) {
            const float pv = s_p[k][n];
            sum += (pv >= m3) ? __expf(pv - m1) : 0.f;
        }
        const float inv = 1.f / sum;
        for (int k = 0; k < KNN; ++k) {
            const float pv = s_p[k][n];
            s_perm[k][n] = (pv >= m3) ? __expf(pv - m1) * inv : 0.f;
        }
    }
    __syncthreads();

    // group-shuffled feature gather: f -> original channel (f%4)*24 + f/4
#pragma unroll
    for (int r = 0; r < 24; ++r) {
        const int pid = tid + 128 * r;
        const int f = pid >> 5, k = pid & 31;
        const int c = (f & 3) * 24 + (f >> 2);
        float val;
        if (c < INC)
            val = feature[(size_t)b * INC * NPTS + (size_t)c * NPTS + s_idx[k]];
        else
            val = s_xf[c - INC][k];
        s_feats[f][k] = val;
    }
    __syncthreads();

    // ---- WMMA f32 16x16x4: g[96,32] = feats[96,32] @ perm[32,32] ----------
    for (int t4 = wave; t4 < 12; t4 += 4) {
        const int mt = t4 >> 1, nt = t4 & 1;
        v8f acc = {0.f, 0.f, 0.f, 0.f, 0.f, 0.f, 0.f, 0.f};
#pragma unroll
        for (int kk = 0; kk < 8; ++kk) {
            const int k0 = kk * 4 + kh * 2;
            v2f a, bb;
            a.x  = s_feats[mt * 16 + lm][k0];
            a.y  = s_feats[mt * 16 + lm][k0 + 1];
            bb.x = s_perm[k0][nt * 16 + lm];
            bb.y = s_perm[k0 + 1][nt * 16 + lm];
            acc = __builtin_amdgcn_wmma_f32_16x16x4_f32(
                      false, a, false, bb, (short)0, acc, false, false);
        }
#pragma unroll
        for (int r = 0; r < 8; ++r)
            s_g[mt * 16 + 8 * kh + r][nt * 16 + lm] = acc[r];
    }
    __syncthreads();

    if (STORE_PHI) {
        // coalesced float4 store of phi row (f-major, n inner = conv_w layout)
        float4* prow = reinterpret_cast<float4*>(phi + (size_t)m * FNN);
#pragma unroll
        for (int q = tid; q < FNN / 4; q += 128) {
            const int kk = q * 4;
            float4 t;
            t.x = s_g[kk >> 5][(kk    ) & 31];
            t.y = s_g[kk >> 5][(kk + 1) & 31];
            t.z = s_g[kk >> 5][(kk + 2) & 31];
            t.w = s_g[kk >> 5][(kk + 3) & 31];
            prow[q] = t;
        }
    } else {
        // fallback: per-point GEMV out[o] = g_flat . conv_w[o,:] + conv_b[o]
        const int o = tid & 63, half = tid >> 6;
        const int base = half * 1536;
        const float4* wrow =
            reinterpret_cast<const float4*>(conv_w + (size_t)o * FNN + base);
        float acc = 0.f;
        for (int q = 0; q < 384; ++q) {
            const float4 w4 = wrow[q];
            const int kk = base + q * 4;
            acc += w4.x * s_g[(kk    ) >> 5][(kk    ) & 31];
            acc += w4.y * s_g[(kk + 1) >> 5][(kk + 1) & 31];
            acc += w4.z * s_g[(kk + 2) >> 5][(kk + 2) & 31];
            acc += w4.w * s_g[(kk + 3) >> 5][(kk + 3) & 31];
        }
        s_red[tid] = acc;
        __syncthreads();
        if (tid < OUTC) {
            const float t = s_red[tid] + s_red[tid + 64] + conv_b[tid];
            out[(size_t)b * OUTC * NPTS + (size_t)tid * NPTS + i] = t;
        }
    }
}

// ---------------------------------------------------------------------------
// Kernel 2b (big-ws path): conv GEMM via WMMA.
//   out_tile[16,64] = phi_tile[16,3072] @ conv_w^T[3072,64]  (+ bias)
//   256 threads = 8 waves = 4 N-tiles x 2 K-halves; 2-way LDS reduction.
//   conv_w is [o][3072] row-major => B fragment loads are contiguous along K.
// ---------------------------------------------------------------------------
__global__ __launch_bounds__(256) void conv_kernel(
    const float* __restrict__ phi, const float* __restrict__ conv_w,
    const float* __restrict__ conv_b, float* __restrict__ out) {

    __shared__ float s_part[8][256];   // 8 KB partial D tiles

    const int tid  = threadIdx.x;
    const int lane = tid & 31;
    const int wave = tid >> 5;
    const int lm   = lane & 15;
    const int kh   = lane >> 4;
    const int nt   = wave & 3;         // output-channel tile
    const int k2   = wave >> 2;        // K half (0..1)

    const float* arow = phi + ((size_t)blockIdx.x * 16 + lm) * FNN;   // point row
    const float* brow = conv_w + (size_t)(nt * 16 + lm) * FNN;        // channel row

    v8f acc = {0.f, 0.f, 0.f, 0.f, 0.f, 0.f, 0.f, 0.f};
#pragma unroll 4
    for (int step = 0; step < 384; ++step) {
        const int k0 = k2 * 1536 + step * 4 + kh * 2;
        const v2f a  = *reinterpret_cast<const v2f*>(arow + k0);
        const v2f bb = *reinterpret_cast<const v2f*>(brow + k0);
        acc = __builtin_amdgcn_wmma_f32_16x16x4_f32(
                  false, a, false, bb, (short)0, acc, false, false);
    }
#pragma unroll
    for (int r = 0; r < 8; ++r)
        s_part[wave][(r + 8 * kh) * 16 + lm] = acc[r];
    __syncthreads();

    for (int q = tid; q < 1024; q += 256) {
        const int p = q >> 6;              // point in tile
        const int o = q & 63;              // output channel
        const int t = o >> 4, col = o & 15;
        const float val = s_part[t][p * 16 + col] + s_part[t + 4][p * 16 + col]
                        + conv_b[o];
        const int mm = blockIdx.x * 16 + p;
        const int bb = mm >> 11, ii = mm & (NPTS - 1);
        out[((size_t)bb * OUTC + o) * NPTS + ii] = val;
    }
}

// ---------------------------------------------------------------------------
// Kernel 3: per-channel batch statistics (mean, biased var over B*N).
// ---------------------------------------------------------------------------
__global__ __launch_bounds__(256) void bn_stats_kernel(const float* __restrict__ out,
                                                       float* __restrict__ stats) {
    __shared__ float s_s[256], s_q[256];
    const int o = blockIdx.x, tid = threadIdx.x;
    float s = 0.f, q = 0.f;
    for (int b = 0; b < BATCH; ++b) {
        const float* p = out + (size_t)b * OUTC * NPTS + (size_t)o * NPTS;
        for (int n = tid; n < NPTS; n += 256) { const float v = p[n]; s += v; q += v * v; }
    }
    s_s[tid] = s; s_q[tid] = q;
    __syncthreads();
    for (int st = 128; st > 0; st >>= 1) {
        if (tid < st) { s_s[tid] += s_s[tid + st]; s_q[tid] += s_q[tid + st]; }
        __syncthreads();
    }
    if (tid == 0) {
        const float inv  = 1.f / (float)(BATCH * NPTS);
        const float mean = s_s[0] * inv;
        const float var  = s_q[0] * inv - mean * mean;
        stats[o] = mean; stats[OUTC + o] = var;
    }
}

// ---------------------------------------------------------------------------
// Kernel 4: apply BatchNorm in place.
// ---------------------------------------------------------------------------
__global__ __launch_bounds__(256) void bn_apply_kernel(
    float* __restrict__ out, const float* __restrict__ stats,
    const float* __restrict__ gamma, const float* __restrict__ beta) {
    const int e = blockIdx.x * 256 + threadIdx.x;
    const int o = (e >> 11) & (OUTC - 1);
    const float mean = stats[o];
    const float inv  = rsqrtf(stats[OUTC + o] + 1e-5f);
    out[e] = gamma[o] * (out[e] - mean) * inv + beta[o];
}

// ---------------------------------------------------------------------------
extern "C" void kernel_launch(void* const* d_in, const int* in_sizes, int n_in,
                              void* d_out, int out_size, void* d_ws, size_t ws_size,
                              hipStream_t stream) {
    const float* x        = (const float*)d_in[0];
    const float* feature  = (const float*)d_in[1];
    const float* B_mat    = (const float*)d_in[2];
    const float* kernels  = (const float*)d_in[3];
    const float* mlp_w    = (const float*)d_in[4];
    const float* mlp_b    = (const float*)d_in[5];
    const float* conv_w   = (const float*)d_in[6];
    const float* conv_b   = (const float*)d_in[7];
    const float* bn_gamma = (const float*)d_in[8];
    const float* bn_beta  = (const float*)d_in[9];
    float* out = (float*)d_out;

    const size_t idx_bytes   = (size_t)MTOT * KNN * sizeof(int);   // 2 MB
    const size_t stats_bytes = 4096;                               // padded
    const size_t phi_bytes   = (size_t)MTOT * FNN * sizeof(float); // 201 MB

    int*   idx   = (int*)d_ws;
    float* stats = (float*)((char*)d_ws + idx_bytes);
    float* phi   = (float*)((char*)d_ws + idx_bytes + stats_bytes);

    knn_kernel<<<MTOT, 256, 0, stream>>>(x, idx);

    if (ws_size >= idx_bytes + stats_bytes + phi_bytes) {
        // big-workspace path: materialize phi, weight-stationary WMMA GEMM
        point_kernel<true><<<MTOT, 128, 0, stream>>>(
            x, feature, B_mat, kernels, mlp_w, mlp_b, conv_w, conv_b,
            idx, phi, out);
        conv_kernel<<<MTOT / 16, 256, 0, stream>>>(phi, conv_w, conv_b, out);
    } else {
        // fallback: fused per-point GEMV
        point_kernel<false><<<MTOT, 128, 0, stream>>>(
            x, feature, B_mat, kernels, mlp_w, mlp_b, conv_w, conv_b,
            idx, phi, out);
    }

    bn_stats_kernel<<<OUTC, 256, 0, stream>>>(out, stats);
    bn_apply_kernel<<<(MTOT * OUTC) / 256, 256, 0, stream>>>(out, stats,
                                                             bn_gamma, bn_beta);
}